// angle_distribution_56521769615648
// MI455X (gfx1250) — compile-verified
//
#include <hip/hip_runtime.h>

#define NBINS 64
#define ANGLE_START (-1.0f)
#define ANGLE_END   (1.0f)
#define LOG2E 1.44269504088896340736f

typedef float v2f __attribute__((ext_vector_type(2)));
typedef float v8f __attribute__((ext_vector_type(8)));

__device__ __forceinline__ float mimage(float d, float c) {
    // minimal image: matches reference _get_offsets semantics
    if (d > 0.5f * c)       d -= c;
    else if (d < -0.5f * c) d += c;
    return d;
}

// ---------------------------------------------------------------------------
// Kernel 1: per-(b,a) cosine of the triplet angle, written to the cos output
// region of d_out (reference output #3, shape [B, A] row-major).
// ---------------------------------------------------------------------------
__global__ __launch_bounds__(256) void cos_kernel(
    const float* __restrict__ xyz, const long long* __restrict__ angle_list,
    const float* __restrict__ cell, float* __restrict__ cos_out,
    int An, int Nn, int total) {
    int idx = blockIdx.x * blockDim.x + threadIdx.x;
    if (idx >= total) return;
    int b = idx / An;
    int a = idx - b * An;
    long long i = angle_list[3ll * a + 0];
    long long j = angle_list[3ll * a + 1];
    long long k = angle_list[3ll * a + 2];
    const float* base = xyz + (size_t)b * Nn * 3;
    const float* pi = base + 3ll * i;
    const float* pj = base + 3ll * j;
    const float* pk = base + 3ll * k;
    float cx = cell[0], cy = cell[1], cz = cell[2];
    float ax = mimage(pi[0] - pj[0], cx);
    float ay = mimage(pi[1] - pj[1], cy);
    float az = mimage(pi[2] - pj[2], cz);
    float bx = mimage(pk[0] - pj[0], cx);
    float by = mimage(pk[1] - pj[1], cy);
    float bz = mimage(pk[2] - pj[2], cz);
    float dot = ax * bx + ay * by + az * bz;
    float n1  = ax * ax + ay * ay + az * az;
    float n2  = bx * bx + by * by + bz * bz;
    cos_out[idx] = dot / sqrtf(n1 * n2);
}

// ---------------------------------------------------------------------------
// Kernel 2a: zero the 64-bin global accumulator in d_ws (ws is poisoned).
// ---------------------------------------------------------------------------
__global__ void zero_kernel(float* __restrict__ hist) {
    hist[threadIdx.x] = 0.0f;
}

// ---------------------------------------------------------------------------
// Kernel 2b: Gaussian-smeared soft histogram, reduced with WMMA.
//
//   count[n] = sum_m exp2( log2e * coeff_n * (cos_m - off_n)^2 )
//
// Per wave: 4 accumulators (v8f) cover 64 bins. A-matrix = ones(16x4),
// B-matrix per WMMA = exp tile of 4 samples x 16 bins. Per the ISA B layout
// for 16x16x4 f32: B VGPR0 = {K=0 rows on lanes 0-15, K=2 on lanes 16-31},
// VGPR1 = {K=1, K=3}; N = lane % 16.  D = 1*B + C accumulates the bin sums
// into every row; at the end row M=0 (acc[g][0], lanes 0-15) is the partial
// histogram for bins [16g, 16g+15], flushed with one f32 atomic per bin.
// EXEC stays all-1s through all WMMAs (uniform loop, full waves).
// ---------------------------------------------------------------------------
__global__ __launch_bounds__(256) void hist_wmma_kernel(
    const float* __restrict__ cos_arr,
    const float* __restrict__ offsets, const float* __restrict__ widths,
    float* __restrict__ hist, int nquads, int total) {
    const int lane = threadIdx.x & 31;
    const int col  = lane & 15;

    float off[4], cf2[4];
#pragma unroll
    for (int g = 0; g < 4; ++g) {
        off[g]  = offsets[col + 16 * g];
        float w = widths[col + 16 * g];
        cf2[g]  = (-0.5f / (w * w)) * LOG2E;   // fold log2(e) into coeff
    }

    v2f amat; amat[0] = 1.0f; amat[1] = 1.0f;  // ones 16x4 A-matrix
    v8f acc[4] = {};

    // lanes 0-15 supply samples {0,1} of each quad, lanes 16-31 samples {2,3}
    const int base_s = (lane < 16) ? 0 : 2;
    const unsigned wave   = (blockIdx.x * blockDim.x + threadIdx.x) >> 5;
    const unsigned nwaves = (gridDim.x * blockDim.x) >> 5;

    for (unsigned q = wave; q < (unsigned)nquads; q += nwaves) {
        float c0 = cos_arr[4u * q + base_s + 0];
        float c1 = cos_arr[4u * q + base_s + 1];
#pragma unroll
        for (int g = 0; g < 4; ++g) {
            float d0 = c0 - off[g];
            float d1 = c1 - off[g];
            v2f bmat;
            bmat[0] = __builtin_amdgcn_exp2f(cf2[g] * d0 * d0);
            bmat[1] = __builtin_amdgcn_exp2f(cf2[g] * d1 * d1);
            acc[g] = __builtin_amdgcn_wmma_f32_16x16x4_f32(
                false, amat, false, bmat, (short)0, acc[g], false, false);
        }
    }

    if (lane < 16) {
#pragma unroll
        for (int g = 0; g < 4; ++g)
            atomicAdd(&hist[16 * g + col], acc[g][0]);
    }

    // scalar tail if total % 4 != 0 (not hit for 4x250000, kept for safety)
    int tail_start = nquads * 4;
    if (wave == 0) {
        for (int m = tail_start + lane; m < total; m += 32) {
            float c = cos_arr[m];
            for (int n = 0; n < NBINS; ++n) {
                float w  = widths[n];
                float cf = (-0.5f / (w * w)) * LOG2E;
                float d  = c - offsets[n];
                atomicAdd(&hist[n], __builtin_amdgcn_exp2f(cf * d * d));
            }
        }
    }
}

// ---------------------------------------------------------------------------
// Kernel 3: normalize count and emit bins linspace. One 64-thread block.
// d_out layout: [bins(65) | count(64) | cos(B*A)]
// ---------------------------------------------------------------------------
__global__ void finalize_kernel(const float* __restrict__ hist,
                                float* __restrict__ out) {
    __shared__ float sh[NBINS];
    __shared__ float total;
    int t = threadIdx.x;  // 0..63
    float v = hist[t];
    sh[t] = v;
    __syncthreads();
    if (t == 0) {
        float s = 0.0f;
        for (int i = 0; i < NBINS; ++i) s += sh[i];
        total = s;
    }
    __syncthreads();
    out[(NBINS + 1) + t] = v / total;                       // count
    float step = (ANGLE_END - ANGLE_START) / (float)NBINS;
    out[t] = ANGLE_START + (float)t * step;                 // bins[0..63]
    if (t == 0) out[NBINS] = ANGLE_END;                     // bins[64]
}

extern "C" void kernel_launch(void* const* d_in, const int* in_sizes, int n_in,
                              void* d_out, int out_size, void* d_ws, size_t ws_size,
                              hipStream_t stream) {
    const float*     xyz        = (const float*)d_in[0];
    const long long* angle_list = (const long long*)d_in[1];  // int64 per reference
    const float*     cell       = (const float*)d_in[2];
    const float*     offsets    = (const float*)d_in[3];
    const float*     widths     = (const float*)d_in[4];
    float*           out        = (float*)d_out;

    const int A     = in_sizes[1] / 3;                       // 250000
    const int BA    = out_size - (NBINS + 1) - NBINS;        // B*A cos values
    const int B     = BA / A;                                // 4
    const int N     = in_sizes[0] / (3 * B);                 // 10000

    float* cos_out = out + (NBINS + 1) + NBINS;
    float* hist    = (float*)d_ws;                           // 64 floats scratch

    cos_kernel<<<(BA + 255) / 256, 256, 0, stream>>>(
        xyz, angle_list, cell, cos_out, A, N, BA);

    zero_kernel<<<1, NBINS, 0, stream>>>(hist);

    hist_wmma_kernel<<<512, 256, 0, stream>>>(
        cos_out, offsets, widths, hist, BA / 4, BA);

    finalize_kernel<<<1, NBINS, 0, stream>>>(hist, out);
}